// AdaptiveSAGE_4337916969114
// MI455X (gfx1250) — compile-verified
//
#include <hip/hip_runtime.h>
#include <stdint.h>

typedef __attribute__((ext_vector_type(2))) float v2f;
typedef __attribute__((ext_vector_type(8))) float v8f;

#define DIM 128

// ---------------------------------------------------------------------------
// Kernel 1: zero workspace (summ[N*128] followed by cnt[N])
// ---------------------------------------------------------------------------
__global__ void zero_ws_kernel(float* __restrict__ ws, long long n) {
    long long i = (long long)blockIdx.x * blockDim.x + threadIdx.x;
    if (i < n) ws[i] = 0.0f;
}

// ---------------------------------------------------------------------------
// Kernel 2: per-edge message + scatter-add (hardware f32 atomics at L2).
// 128 threads handle one edge (one per feature dim) -> coalesced 512B read
// of h[src] and coalesced atomic adds into summ[dst]. summ (25.6 MB) lives
// comfortably in the 192 MB L2, so the atomics resolve there.
// ---------------------------------------------------------------------------
__global__ void edge_scatter_kernel(const float* __restrict__ h,
                                    const float* __restrict__ alpha,
                                    const float* __restrict__ ew,
                                    const int*   __restrict__ node_id,
                                    const int*   __restrict__ esrc,
                                    const int*   __restrict__ edst,
                                    float* __restrict__ summ,
                                    float* __restrict__ cnt,
                                    int E, int gene_num) {
    int e = blockIdx.x * 2 + (threadIdx.x >> 7);
    if (e >= E) return;
    int d   = threadIdx.x & (DIM - 1);
    int src = esrc[e];
    int dst = edst[e];
    int sid = node_id[src];
    int did = node_id[dst];
    // mutually exclusive cases, same outcome as the reference's where-chain
    int idx = gene_num + 1;                        // cell -> cell self-loop
    if      (sid >= 0 && did <  0) idx = sid;      // gene -> cell
    else if (did >= 0 && sid <  0) idx = did;      // cell -> gene
    else if (did >= 0 && sid >= 0) idx = gene_num; // gene -> gene
    float coeff = alpha[idx] * ew[e];
    float hv = h[(size_t)src * DIM + d];
    unsafeAtomicAdd(&summ[(size_t)dst * DIM + d], hv * coeff);
    if (d == 0) unsafeAtomicAdd(&cnt[dst], 1.0f);
}

// ---------------------------------------------------------------------------
// Kernel 3: z = relu((summ/max(cnt,1)) @ W^T + b) via V_WMMA_F32_16X16X4_F32.
//
// Fragment layouts (CDNA5 ISA 7.12.2, wave32):
//   A (16x4 f32, 2 VGPRs):  lane l -> M = l%16, K(v) = v + 2*(l/16)
//   B (4x16 f32, 2 VGPRs):  lane l -> N = l%16, K(v) = v + 2*(l/16)
//   C/D (16x16 f32, 8 VGPRs): VGPR r, lane l -> M = r + 8*(l/16), N = l%16
//
// One wave owns a 16-row M-tile and all 8 column tiles (acc: 64 VGPRs),
// looping K = 128 in steps of 4 -> 8 WMMA per K-step.
//
// W (64 KB of the WGP's 320 KB LDS) is staged once per 256-thread block with
// CDNA5 async copies: GLOBAL_LOAD_ASYNC_TO_LDS_B128 (ASYNCcnt-tracked, no
// VGPR round-trip), then s_wait_asynccnt 0 per wave before the barrier.
// ---------------------------------------------------------------------------
__global__ void __launch_bounds__(256)
gemm_relu_kernel(const float* __restrict__ summ,
                 const float* __restrict__ cnt,
                 const float* __restrict__ W,
                 const float* __restrict__ bias,
                 float* __restrict__ out,
                 int n_nodes, int n_tiles_m) {
    __shared__ float Wlds[DIM * DIM];   // 64 KB

    // async 128x128 f32 stage: 4096 x 16B chunks across 256 lanes (16 each)
    {
        const int tid = threadIdx.x;
        for (int i = tid; i < (DIM * DIM) / 4; i += 256) {
            // low 32 bits of a generic __shared__ pointer == LDS byte offset
            uint32_t lds_addr = (uint32_t)(uintptr_t)(&Wlds[i * 4]);
            const float* gsrc = W + (size_t)i * 4;
            asm volatile("global_load_async_to_lds_b128 %0, %1, off"
                         :: "v"(lds_addr), "v"(gsrc)
                         : "memory");
        }
        asm volatile("s_wait_asynccnt 0x0" ::: "memory");
    }
    __syncthreads();

    const int wave = threadIdx.x >> 5;
    const int lane = threadIdx.x & 31;
    const int tile_m = blockIdx.x * 8 + wave;
    if (tile_m >= n_tiles_m) return;    // wave-uniform -> EXEC stays all-1s

    const int node_base = tile_m * 16;
    const int mrow  = lane & 15;        // M for A-frag, N for B-frag
    const int khalf = (lane >> 4) * 2;  // per-lane K offset within a K=4 step

    int arow_node = node_base + mrow;
    if (arow_node >= n_nodes) arow_node = n_nodes - 1;   // clamp (stores guarded)
    const float inv = 1.0f / fmaxf(cnt[arow_node], 1.0f);
    const float* __restrict__ arow = summ + (size_t)arow_node * DIM + khalf;

    v8f acc[8] = {};

    for (int kb = 0; kb < DIM; kb += 4) {
        v2f a = *(const v2f*)(arow + kb);
        a = a * inv;                    // fold the scatter-mean into A
#pragma unroll
        for (int jt = 0; jt < 8; ++jt) {
            v2f bf = *(const v2f*)(&Wlds[(jt * 16 + mrow) * DIM + kb + khalf]);
            acc[jt] = __builtin_amdgcn_wmma_f32_16x16x4_f32(
                /*neg_a=*/false, a, /*neg_b=*/false, bf,
                /*c_mod=*/(short)0, acc[jt],
                /*reuse_a=*/false, /*reuse_b=*/false);
        }
    }

    // epilogue: bias + ReLU, D layout -> row = r + 8*(lane/16), col = lane%16
#pragma unroll
    for (int jt = 0; jt < 8; ++jt) {
        const int j  = jt * 16 + mrow;
        const float bj = bias[j];
#pragma unroll
        for (int r = 0; r < 8; ++r) {
            const int row = node_base + r + 8 * (lane >> 4);
            if (row < n_nodes) {
                float v = acc[jt][r] + bj;
                out[(size_t)row * DIM + j] = fmaxf(v, 0.0f);
            }
        }
    }
}

// ---------------------------------------------------------------------------
// Launch: zero -> scatter -> WMMA GEMM (all on `stream`, graph-capture safe)
// ---------------------------------------------------------------------------
extern "C" void kernel_launch(void* const* d_in, const int* in_sizes, int n_in,
                              void* d_out, int out_size, void* d_ws, size_t ws_size,
                              hipStream_t stream) {
    const float* h       = (const float*)d_in[0];
    const float* alpha   = (const float*)d_in[1];
    const float* ew      = (const float*)d_in[2];
    const float* W       = (const float*)d_in[3];
    const float* bias    = (const float*)d_in[4];
    const int*   node_id = (const int*)d_in[5];
    const int*   esrc    = (const int*)d_in[6];
    const int*   edst    = (const int*)d_in[7];
    float*       out     = (float*)d_out;

    const int N        = in_sizes[0] / DIM;   // 50000
    const int gene_num = in_sizes[1] - 2;     // 3000
    const int E        = in_sizes[7];         // 1000000

    float* summ = (float*)d_ws;               // [N*128]
    float* cnt  = summ + (size_t)N * DIM;     // [N]

    const long long zn = (long long)N * DIM + N;
    zero_ws_kernel<<<(unsigned)((zn + 255) / 256), 256, 0, stream>>>(summ, zn);

    const int eblocks = (E + 1) / 2;          // 2 edges per 256-thread block
    edge_scatter_kernel<<<eblocks, 256, 0, stream>>>(h, alpha, ew, node_id,
                                                     esrc, edst, summ, cnt,
                                                     E, gene_num);

    const int tiles_m = (N + 15) / 16;        // 3125
    const int gblocks = (tiles_m + 7) / 8;    // 8 waves (M-tiles) per block
    gemm_relu_kernel<<<gblocks, 256, 0, stream>>>(summ, cnt, W, bias, out,
                                                  N, tiles_m);
}